// MLPPredictor_54640573940279
// MI455X (gfx1250) — compile-verified
//
#include <hip/hip_runtime.h>
#include <hip/hip_bf16.h>

typedef float v2f __attribute__((ext_vector_type(2)));
typedef float v8f __attribute__((ext_vector_type(8)));

#define IN_FEATS 128
#define HIDDEN   32
#define N_NODES  40000
#define NODE_TILES (N_NODES / 16)   // 2500, exact

// ---------------------------------------------------------------------------
// Phase 1: Z[n, j] = sum_k h[n,k] * W1[k,j]   (40000x128 @ 128x32, pure f32)
// One wave per 16-node tile. Two 16x16 f32 accumulators cover N=0..31.
// v_wmma_f32_16x16x4_f32, 32 K-steps.
// ---------------------------------------------------------------------------
__global__ void __launch_bounds__(256)
node_gemm_wmma(const float* __restrict__ h,
               const float* __restrict__ W1,
               float* __restrict__ Z) {
    const int lane = threadIdx.x & 31;
    const int wave = threadIdx.x >> 5;
    const int tile = blockIdx.x * (blockDim.x >> 5) + wave;
    if (tile >= NODE_TILES) return;          // wave-uniform: EXEC stays all-1s

    const int mbase = tile * 16;
    const int mrow  = lane & 15;             // A: M index / B: N index
    const int koff  = (lane >> 4) * 2;       // lanes 16-31 hold K+2, K+3

    const float* __restrict__ hrow = h + (size_t)(mbase + mrow) * IN_FEATS;

    v8f acc0 = {};   // N = 0..15
    v8f acc1 = {};   // N = 16..31

    for (int k = 0; k < IN_FEATS; k += 4) {
        // A operand: 16x4 f32 tile of h
        v2f a;
        a.x = hrow[k + koff];
        a.y = hrow[k + koff + 1];
        // B operands: 4x16 f32 tiles of W1 (row-major [128,32])
        v2f b0, b1v;
        b0.x  = W1[(k + koff)     * HIDDEN + mrow];
        b0.y  = W1[(k + koff + 1) * HIDDEN + mrow];
        b1v.x = W1[(k + koff)     * HIDDEN + 16 + mrow];
        b1v.y = W1[(k + koff + 1) * HIDDEN + 16 + mrow];

        acc0 = __builtin_amdgcn_wmma_f32_16x16x4_f32(
                   false, a, false, b0, (short)0, acc0, false, false);
        acc1 = __builtin_amdgcn_wmma_f32_16x16x4_f32(
                   false, a, false, b1v, (short)0, acc1, false, false);
    }

    // D layout: VGPR r -> M = r + (lane>=16 ? 8 : 0), N = lane&15
    const int n    = lane & 15;
    const int madd = (lane >> 4) * 8;
#pragma unroll
    for (int r = 0; r < 8; ++r) {
        float* zrow = Z + (size_t)(mbase + madd + r) * HIDDEN;
        zrow[n]      = acc0[r];
        zrow[16 + n] = acc1[r];
    }
}

// ---------------------------------------------------------------------------
// Phase 2: per edge e: out[e] = relu(Z[src]+Z[dst]+b1) . W2 + b2
// Z (5.1 MB) is L2-resident; 256 B gathered per edge vs 1024 B in reference.
// ---------------------------------------------------------------------------
__global__ void __launch_bounds__(256)
edge_mlp(const float* __restrict__ Z,
         const int*   __restrict__ src,
         const int*   __restrict__ dst,
         const float* __restrict__ b1,
         const float* __restrict__ W2,
         const float* __restrict__ b2,
         float* __restrict__ out,
         int nEdges) {
    const int e = blockIdx.x * blockDim.x + threadIdx.x;
    if (e >= nEdges) return;

    const float4* __restrict__ zs = (const float4*)(Z + (size_t)src[e] * HIDDEN);
    const float4* __restrict__ zd = (const float4*)(Z + (size_t)dst[e] * HIDDEN);
    const float4* __restrict__ B1 = (const float4*)b1;   // uniform -> scalar loads
    const float4* __restrict__ w2 = (const float4*)W2;   // uniform -> scalar loads

    float acc = 0.0f;
#pragma unroll
    for (int i = 0; i < HIDDEN / 4; ++i) {
        const float4 s = zs[i];
        const float4 d = zd[i];
        const float4 bb = B1[i];
        const float4 w  = w2[i];
        acc = fmaf(fmaxf(s.x + d.x + bb.x, 0.0f), w.x, acc);
        acc = fmaf(fmaxf(s.y + d.y + bb.y, 0.0f), w.y, acc);
        acc = fmaf(fmaxf(s.z + d.z + bb.z, 0.0f), w.z, acc);
        acc = fmaf(fmaxf(s.w + d.w + bb.w, 0.0f), w.w, acc);
    }
    out[e] = acc + b2[0];
}

// ---------------------------------------------------------------------------
// Launch
// ---------------------------------------------------------------------------
extern "C" void kernel_launch(void* const* d_in, const int* in_sizes, int n_in,
                              void* d_out, int out_size, void* d_ws, size_t ws_size,
                              hipStream_t stream) {
    const float* h   = (const float*)d_in[0];
    const int*   src = (const int*)  d_in[1];
    const int*   dst = (const int*)  d_in[2];
    const float* W1  = (const float*)d_in[3];
    const float* b1  = (const float*)d_in[4];
    const float* W2  = (const float*)d_in[5];
    const float* b2  = (const float*)d_in[6];
    float*       out = (float*)d_out;

    float* Z = (float*)d_ws;                 // 40000 * 32 * 4 = 5.12 MB scratch
    const int nEdges = in_sizes[1];

    // Phase 1: 2500 wave-tiles, 8 waves per 256-thread block -> 313 blocks
    const int wavesPerBlock = 256 / 32;
    const int gemmBlocks = (NODE_TILES + wavesPerBlock - 1) / wavesPerBlock;
    node_gemm_wmma<<<gemmBlocks, 256, 0, stream>>>(h, W1, Z);

    // Phase 2: one thread per edge
    const int edgeBlocks = (nEdges + 255) / 256;
    edge_mlp<<<edgeBlocks, 256, 0, stream>>>(Z, src, dst, b1, W2, b2, out, nEdges);
}